// LayoutModel_72018011619522
// MI455X (gfx1250) — compile-verified
//
#include <hip/hip_runtime.h>
#include <math.h>

typedef float v2f __attribute__((ext_vector_type(2)));
typedef float v8f __attribute__((ext_vector_type(8)));
typedef unsigned int u32x4 __attribute__((ext_vector_type(4)));
typedef int i32x8 __attribute__((ext_vector_type(8)));
typedef int i32x4 __attribute__((ext_vector_type(4)));

#define HID 128
#define NFD 140
#define OPCD 64
#define NOPC 120
#define CFGD 18
#define CHUNK 512

__device__ __forceinline__ float gelu_f(float x) {
    return 0.5f * x * (1.0f + erff(x * 0.70710678118654752f));
}
__device__ __forceinline__ float sigmoid_f(float x) {
    return 1.0f / (1.0f + expf(-x));
}

// ---------------------------------------------------------------------------
// Tensor Data Mover: 2D tile (width x height, 4-byte elems) global -> LDS.
// Descriptor packing per CDNA5 ISA 8.3/8.4 (type=2, data_size=4B, 2D tensor,
// groups 2/3 zero). Issued by a single lane; TDM ignores EXEC, one issue/wave.
// ---------------------------------------------------------------------------
__device__ __forceinline__ void tdm_load_2d(unsigned lds_off, const void* gptr,
                                            unsigned width, unsigned height,
                                            unsigned stride) {
    unsigned long long ga = (unsigned long long)gptr;
    u32x4 g0;
    g0[0] = 1u;                                   // count=1, is_restore=0, gather=0
    g0[1] = lds_off;                              // lds_addr (bytes)
    g0[2] = (unsigned)ga;                         // global_addr[31:0]
    g0[3] = (unsigned)(ga >> 32) | 0x80000000u;   // global_addr[56:32] | type=2
    i32x8 g1;
    g1[0] = 0x20000;                              // data_size=2 (4B); mask/flags=0
    g1[1] = (int)((width & 0xFFFFu) << 16);       // tensor_dim0 lo16 (abar_addr=0)
    g1[2] = (int)((width >> 16) | ((height & 0xFFFFu) << 16));   // dim0 hi | dim1 lo
    g1[3] = (int)((height >> 16) | ((width & 0xFFFFu) << 16));   // dim1 hi | tile_dim0
    g1[4] = (int)(height & 0xFFFFu);              // tile_dim1 (tile_dim2=0)
    g1[5] = (int)stride;                          // tensor_dim0_stride[31:0]
    g1[6] = 0;                                    // stride hi16 | dim1_stride lo16
    g1[7] = 0;                                    // dim1_stride hi32
    i32x4 z4 = {0, 0, 0, 0};
    i32x8 z8 = {0, 0, 0, 0, 0, 0, 0, 0};
    __builtin_amdgcn_tensor_load_to_lds(g0, g1, z4, z4, z8, 0);
}

__device__ __forceinline__ unsigned lds_off_of(const void* p) {
    // generic pointers to LDS carry the 32-bit LDS offset in the low dword
    return (unsigned)(unsigned long long)p;
}

// ---------------------------------------------------------------------------
// opc_proj[o][j] = sum_k opcode_embed[o][k] * in_w[140+k][j]
// ---------------------------------------------------------------------------
__global__ void __launch_bounds__(128) k_opc_proj(const float* __restrict__ emb,
                                                  const float* __restrict__ in_w,
                                                  float* __restrict__ opc_proj) {
    int o = blockIdx.x, t = threadIdx.x;
    float acc = 0.f;
#pragma unroll 8
    for (int k = 0; k < OPCD; ++k)
        acc += emb[o * OPCD + k] * in_w[(NFD + k) * HID + t];
    opc_proj[o * HID + t] = acc;
}

// ---------------------------------------------------------------------------
// Embed GEMM: conv[n][j] = sum_{k<140} node_feat[n][k] * in_w[k][j]
// WMMA f32 16x16x4; weights (140x128 = 71.7KB) staged in LDS via TDM.
// ---------------------------------------------------------------------------
__global__ void __launch_bounds__(256) k_embed_gemm(const float* __restrict__ nf,
                                                    const float* __restrict__ in_w,
                                                    float* __restrict__ out, int N) {
    __shared__ float wlds[NFD * HID];
    const int t = threadIdx.x;
    const int lane = t & 31, wave = t >> 5;
    const int tile = blockIdx.x * 8 + wave;
    const bool active = (tile * 16 < N);
    const int row0 = tile * 16;
    const int m = lane & 15, pair = lane >> 4;

    if (t == 0) {
        tdm_load_2d(lds_off_of(wlds), in_w, HID, NFD, HID);
        __builtin_amdgcn_s_wait_tensorcnt(0);
    }
    __syncthreads();

    v8f acc[8];
#pragma unroll
    for (int jt = 0; jt < 8; ++jt) acc[jt] = (v8f){0, 0, 0, 0, 0, 0, 0, 0};

    if (active) {
        const float* arow = nf + (size_t)(row0 + m) * NFD;
        for (int k = 0; k < NFD; k += 4) {
            v2f a;
            a.x = arow[k + 2 * pair];
            a.y = arow[k + 2 * pair + 1];
#pragma unroll
            for (int jt = 0; jt < 8; ++jt) {
                v2f b;
                b.x = wlds[(k + 2 * pair) * HID + jt * 16 + m];
                b.y = wlds[(k + 2 * pair + 1) * HID + jt * 16 + m];
                acc[jt] = __builtin_amdgcn_wmma_f32_16x16x4_f32(false, a, false, b,
                                                                (short)0, acc[jt], false, false);
            }
        }
#pragma unroll
        for (int jt = 0; jt < 8; ++jt)
#pragma unroll
            for (int v = 0; v < 8; ++v)
                out[(size_t)(row0 + v + 8 * pair) * HID + jt * 16 + m] = acc[jt][v];
    }
}

// ---------------------------------------------------------------------------
// Embed post: h = gelu(LN(conv + opc_proj[opc] + topo*w204 + in_b))  (wave/node)
// ---------------------------------------------------------------------------
__global__ void __launch_bounds__(256) k_embed_post(const float* __restrict__ conv,
                                                    const int* __restrict__ opcode,
                                                    const float* __restrict__ topo,
                                                    const float* __restrict__ opc_proj,
                                                    const float* __restrict__ in_w,
                                                    const float* __restrict__ in_b,
                                                    const float* __restrict__ in_g,
                                                    const float* __restrict__ in_beta,
                                                    float* __restrict__ h, int N) {
    int lane = threadIdx.x & 31;
    int node = (blockIdx.x * blockDim.x + threadIdx.x) >> 5;
    if (node >= N) return;
    int opc = opcode[node];
    opc = opc < 0 ? 0 : (opc > NOPC - 1 ? NOPC - 1 : opc);
    float td = topo[node];
    float4 cv = ((const float4*)(conv + (size_t)node * HID))[lane];
    float4 ov = ((const float4*)(opc_proj + (size_t)opc * HID))[lane];
    float4 wv = ((const float4*)(in_w + 204 * HID))[lane];
    float4 bv = ((const float4*)in_b)[lane];
    float x0 = cv.x + ov.x + td * wv.x + bv.x;
    float x1 = cv.y + ov.y + td * wv.y + bv.y;
    float x2 = cv.z + ov.z + td * wv.z + bv.z;
    float x3 = cv.w + ov.w + td * wv.w + bv.w;
    float s = x0 + x1 + x2 + x3;
#pragma unroll
    for (int off = 16; off >= 1; off >>= 1) s += __shfl_xor(s, off, 32);
    float mu = s * (1.0f / HID);
    float d0 = x0 - mu, d1 = x1 - mu, d2 = x2 - mu, d3 = x3 - mu;
    float q = d0 * d0 + d1 * d1 + d2 * d2 + d3 * d3;
#pragma unroll
    for (int off = 16; off >= 1; off >>= 1) q += __shfl_xor(q, off, 32);
    float r = rsqrtf(q * (1.0f / HID) + 1e-5f);
    float4 g = ((const float4*)in_g)[lane];
    float4 be = ((const float4*)in_beta)[lane];
    float4 y;
    y.x = gelu_f(d0 * r * g.x + be.x);
    y.y = gelu_f(d1 * r * g.y + be.y);
    y.z = gelu_f(d2 * r * g.z + be.z);
    y.w = gelu_f(d3 * r * g.w + be.w);
    ((float4*)(h + (size_t)node * HID))[lane] = y;
}

// ---------------------------------------------------------------------------
// CSR build
// ---------------------------------------------------------------------------
__global__ void __launch_bounds__(256) k_deg(const int* __restrict__ ei, unsigned* __restrict__ deg,
                                             int M) {
    int e = blockIdx.x * blockDim.x + threadIdx.x;
    if (e < M) atomicAdd(&deg[ei[M + e]], 1u);
}

__global__ void __launch_bounds__(1024) k_scan(const unsigned* __restrict__ deg,
                                               unsigned* __restrict__ ptr, int N) {
    __shared__ unsigned sh[1024];
    unsigned carry = 0;
    for (int base = 0; base < N; base += 1024) {
        int i = base + (int)threadIdx.x;
        unsigned v = (i < N) ? deg[i] : 0u;
        sh[threadIdx.x] = v;
        __syncthreads();
        for (int off = 1; off < 1024; off <<= 1) {
            unsigned t = (threadIdx.x >= (unsigned)off) ? sh[threadIdx.x - off] : 0u;
            __syncthreads();
            sh[threadIdx.x] += t;
            __syncthreads();
        }
        if (i < N) ptr[i] = carry + sh[threadIdx.x] - v;  // exclusive
        unsigned total = sh[1023];
        __syncthreads();
        carry += total;
    }
    if (threadIdx.x == 0) ptr[N] = carry;
}

__global__ void __launch_bounds__(256) k_fill(const int* __restrict__ ei,
                                              const unsigned* __restrict__ ptr,
                                              unsigned* __restrict__ cursor,
                                              int* __restrict__ csr_src, int M) {
    int e = blockIdx.x * blockDim.x + threadIdx.x;
    if (e < M) {
        int d = ei[M + e];
        unsigned slot = atomicAdd(&cursor[d], 1u);
        csr_src[ptr[d] + slot] = ei[e];
    }
}

// ---------------------------------------------------------------------------
// Mean aggregation: agg[d] = (1/max(deg,1)) * sum_{e in in(d)} h[src[e]]
// one wave per destination node, float4 per lane.
// ---------------------------------------------------------------------------
__global__ void __launch_bounds__(256) k_agg(const float* __restrict__ h,
                                             const unsigned* __restrict__ ptr,
                                             const int* __restrict__ csr_src,
                                             float* __restrict__ agg, int N) {
    int lane = threadIdx.x & 31;
    int node = (blockIdx.x * blockDim.x + threadIdx.x) >> 5;
    if (node >= N) return;
    unsigned s0 = ptr[node], s1 = ptr[node + 1];
    float4 acc = {0.f, 0.f, 0.f, 0.f};
    for (unsigned e = s0; e < s1; ++e) {
        int src = csr_src[e];
        if (e + 1 < s1) __builtin_prefetch(h + (size_t)csr_src[e + 1] * HID, 0, 0);
        float4 v = ((const float4*)(h + (size_t)src * HID))[lane];
        acc.x += v.x; acc.y += v.y; acc.z += v.z; acc.w += v.w;
    }
    unsigned dg = s1 - s0;
    float inv = 1.0f / (float)(dg < 1u ? 1u : dg);
    acc.x *= inv; acc.y *= inv; acc.z *= inv; acc.w *= inv;
    ((float4*)(agg + (size_t)node * HID))[lane] = acc;
}

// ---------------------------------------------------------------------------
// SAGE GEMM: conv[n] = agg[n] @ Wl + h[n] @ Wr   (WMMA f32)
// Both weight tiles DMA'd up-front into double-buffered LDS via TDM; the Wr
// transfer overlaps the entire agg@Wl pass (TDM ops complete in order, so
// tensorcnt<=1 gates pass 1, tensorcnt==0 gates pass 2).
// conv aliases agg: each wave reads only its own rows before storing.
// ---------------------------------------------------------------------------
__global__ void __launch_bounds__(256) k_sage_gemm(const float* __restrict__ agg,
                                                   const float* __restrict__ h,
                                                   const float* __restrict__ Wl,
                                                   const float* __restrict__ Wr,
                                                   float* __restrict__ conv, int N) {
    __shared__ float wl_lds[HID * HID];  // 64 KB
    __shared__ float wr_lds[HID * HID];  // 64 KB
    const int t = threadIdx.x;
    const int lane = t & 31, wave = t >> 5;
    const int tile = blockIdx.x * 8 + wave;
    const bool active = (tile * 16 < N);
    const int row0 = tile * 16;
    const int m = lane & 15, pair = lane >> 4;

    if (t == 0) {
        tdm_load_2d(lds_off_of(wl_lds), Wl, HID, HID, HID);
        tdm_load_2d(lds_off_of(wr_lds), Wr, HID, HID, HID);
        __builtin_amdgcn_s_wait_tensorcnt(1);  // Wl resident; Wr still in flight
    }
    __syncthreads();

    v8f acc[8];
#pragma unroll
    for (int jt = 0; jt < 8; ++jt) acc[jt] = (v8f){0, 0, 0, 0, 0, 0, 0, 0};

    if (active) {
        const float* arow = agg + (size_t)(row0 + m) * HID;
        for (int k = 0; k < HID; k += 4) {
            v2f a;
            a.x = arow[k + 2 * pair];
            a.y = arow[k + 2 * pair + 1];
#pragma unroll
            for (int jt = 0; jt < 8; ++jt) {
                v2f b;
                b.x = wl_lds[(k + 2 * pair) * HID + jt * 16 + m];
                b.y = wl_lds[(k + 2 * pair + 1) * HID + jt * 16 + m];
                acc[jt] = __builtin_amdgcn_wmma_f32_16x16x4_f32(false, a, false, b,
                                                                (short)0, acc[jt], false, false);
            }
        }
    }
    if (t == 0) __builtin_amdgcn_s_wait_tensorcnt(0);  // Wr resident
    __syncthreads();
    if (active) {
        const float* hrow = h + (size_t)(row0 + m) * HID;
        for (int k = 0; k < HID; k += 4) {
            v2f a;
            a.x = hrow[k + 2 * pair];
            a.y = hrow[k + 2 * pair + 1];
#pragma unroll
            for (int jt = 0; jt < 8; ++jt) {
                v2f b;
                b.x = wr_lds[(k + 2 * pair) * HID + jt * 16 + m];
                b.y = wr_lds[(k + 2 * pair + 1) * HID + jt * 16 + m];
                acc[jt] = __builtin_amdgcn_wmma_f32_16x16x4_f32(false, a, false, b,
                                                                (short)0, acc[jt], false, false);
            }
        }
#pragma unroll
        for (int jt = 0; jt < 8; ++jt)
#pragma unroll
            for (int v = 0; v < 8; ++v)
                conv[(size_t)(row0 + v + 8 * pair) * HID + jt * 16 + m] = acc[jt][v];
    }
}

// ---------------------------------------------------------------------------
// SAGE post: h = LN(h + gelu(conv + bl), g, b)  (wave/node)
// ---------------------------------------------------------------------------
__global__ void __launch_bounds__(256) k_sage_post(float* __restrict__ h,
                                                   const float* __restrict__ conv,
                                                   const float* __restrict__ bl,
                                                   const float* __restrict__ g,
                                                   const float* __restrict__ b, int N) {
    int lane = threadIdx.x & 31;
    int node = (blockIdx.x * blockDim.x + threadIdx.x) >> 5;
    if (node >= N) return;
    float4 hv = ((const float4*)(h + (size_t)node * HID))[lane];
    float4 cv = ((const float4*)(conv + (size_t)node * HID))[lane];
    float4 blv = ((const float4*)bl)[lane];
    float x0 = hv.x + gelu_f(cv.x + blv.x);
    float x1 = hv.y + gelu_f(cv.y + blv.y);
    float x2 = hv.z + gelu_f(cv.z + blv.z);
    float x3 = hv.w + gelu_f(cv.w + blv.w);
    float s = x0 + x1 + x2 + x3;
#pragma unroll
    for (int off = 16; off >= 1; off >>= 1) s += __shfl_xor(s, off, 32);
    float mu = s * (1.0f / HID);
    float d0 = x0 - mu, d1 = x1 - mu, d2 = x2 - mu, d3 = x3 - mu;
    float q = d0 * d0 + d1 * d1 + d2 * d2 + d3 * d3;
#pragma unroll
    for (int off = 16; off >= 1; off >>= 1) q += __shfl_xor(q, off, 32);
    float r = rsqrtf(q * (1.0f / HID) + 1e-5f);
    float4 gv = ((const float4*)g)[lane];
    float4 bv = ((const float4*)b)[lane];
    float4 y;
    y.x = d0 * r * gv.x + bv.x;
    y.y = d1 * r * gv.y + bv.y;
    y.z = d2 * r * gv.z + bv.z;
    y.w = d3 * r * gv.w + bv.w;
    ((float4*)(h + (size_t)node * HID))[lane] = y;
}

// ---------------------------------------------------------------------------
// gsum[j] += partial column sums of h  (for global_embed mean)
// ---------------------------------------------------------------------------
__global__ void __launch_bounds__(128) k_mean(const float* __restrict__ h,
                                              float* __restrict__ gsum, int N) {
    int t = threadIdx.x;
    float acc = 0.f;
    for (int n = blockIdx.x; n < N; n += gridDim.x) acc += h[(size_t)n * HID + t];
    atomicAdd(&gsum[t], acc);
}

// ---------------------------------------------------------------------------
// base[nc][j] = h[ids[nc]] @ cfg_w[:128] + cfg_b
// ---------------------------------------------------------------------------
__global__ void __launch_bounds__(128) k_base(const float* __restrict__ h,
                                              const int* __restrict__ ids,
                                              const float* __restrict__ cfg_w,
                                              const float* __restrict__ cfg_b,
                                              float* __restrict__ base) {
    __shared__ float hr[HID];
    int nc = blockIdx.x, t = threadIdx.x;
    int nid = ids[nc];
    hr[t] = h[(size_t)nid * HID + t];
    __syncthreads();
    float acc = cfg_b[t];
#pragma unroll 8
    for (int k = 0; k < HID; ++k) acc += hr[k] * cfg_w[k * HID + t];
    base[nc * HID + t] = acc;
}

// ---------------------------------------------------------------------------
// Squeeze-excite helper (t = hidden index, 128 threads). 3 barriers.
// ---------------------------------------------------------------------------
__device__ __forceinline__ float se_apply(float p, int t, const float* w1r, const float* w2c,
                                          float b1t, float b2t, float* psh, float* part,
                                          float* s1) {
    int j = t & 15, g = t >> 4;
    psh[t] = p;
    __syncthreads();
    float sp = 0.f;
#pragma unroll
    for (int r = 0; r < 16; ++r) sp += psh[g * 16 + r] * w1r[r];
    part[t] = sp;
    __syncthreads();
    if (t < 16) {
        float s = b1t;
#pragma unroll
        for (int q = 0; q < 8; ++q) s += part[q * 16 + t];
        s1[t] = fmaxf(s, 0.f);
    }
    __syncthreads();
    float sc = b2t;
#pragma unroll
    for (int r = 0; r < 16; ++r) sc += s1[r] * w2c[r];
    return p * sigmoid_f(sc);
}

// ---------------------------------------------------------------------------
// Pass 1: per-nc block, accumulate sumexp[nc][chunk][j] over configs.
// ---------------------------------------------------------------------------
__global__ void __launch_bounds__(128) k_pass1(const float* __restrict__ base,
                                               const float* __restrict__ cfg_feat,
                                               const float* __restrict__ cfg_w,
                                               const float* __restrict__ se_w1,
                                               const float* __restrict__ se_b1,
                                               const float* __restrict__ se_w2,
                                               const float* __restrict__ se_b2,
                                               const float* __restrict__ temp_p,
                                               float* __restrict__ sumexp, int C, int NC) {
    __shared__ float psh[HID], part[HID], s1[16];
    int nc = blockIdx.x, t = threadIdx.x;
    int j = t & 15, g = t >> 4;
    float invT = 1.0f / temp_p[0];
    float bs = base[nc * HID + t];
    float wc[CFGD];
#pragma unroll
    for (int k = 0; k < CFGD; ++k) wc[k] = cfg_w[(HID + k) * HID + t];
    float w1r[16], w2c[16];
#pragma unroll
    for (int r = 0; r < 16; ++r) w1r[r] = se_w1[(g * 16 + r) * 16 + j];
#pragma unroll
    for (int r = 0; r < 16; ++r) w2c[r] = se_w2[r * HID + t];
    float b1t = (t < 16) ? se_b1[t] : 0.f;
    float b2t = se_b2[t];
    float acc0 = 0.f, acc1 = 0.f;
    for (int c = 0; c < C; ++c) {
        const float* cf = cfg_feat + ((size_t)c * NC + nc) * CFGD;
        float p = bs;
#pragma unroll
        for (int k = 0; k < CFGD; ++k) p += cf[k] * wc[k];
        p = gelu_f(p);
        p = se_apply(p, t, w1r, w2c, b1t, b2t, psh, part, s1);
        float e = expf(p * invT);
        if (c < CHUNK) acc0 += e; else acc1 += e;
        __syncthreads();
    }
    sumexp[(nc * 2 + 0) * HID + t] = acc0;
    sumexp[(nc * 2 + 1) * HID + t] = acc1;
}

// ---------------------------------------------------------------------------
// Pass 2: per-config block, recompute p, weight by softmax, mean over nc.
// ---------------------------------------------------------------------------
__global__ void __launch_bounds__(128) k_pass2(const float* __restrict__ base,
                                               const float* __restrict__ cfg_feat,
                                               const float* __restrict__ cfg_w,
                                               const float* __restrict__ se_w1,
                                               const float* __restrict__ se_b1,
                                               const float* __restrict__ se_w2,
                                               const float* __restrict__ se_b2,
                                               const float* __restrict__ sumexp,
                                               const float* __restrict__ temp_p,
                                               float* __restrict__ cfge, int C, int NC) {
    __shared__ float psh[HID], part[HID], s1[16];
    int c = blockIdx.x, t = threadIdx.x;
    int j = t & 15, g = t >> 4;
    int chunk = c / CHUNK;
    float invT = 1.0f / temp_p[0];
    float wc[CFGD];
#pragma unroll
    for (int k = 0; k < CFGD; ++k) wc[k] = cfg_w[(HID + k) * HID + t];
    float w1r[16], w2c[16];
#pragma unroll
    for (int r = 0; r < 16; ++r) w1r[r] = se_w1[(g * 16 + r) * 16 + j];
#pragma unroll
    for (int r = 0; r < 16; ++r) w2c[r] = se_w2[r * HID + t];
    float b1t = (t < 16) ? se_b1[t] : 0.f;
    float b2t = se_b2[t];
    float acc = 0.f;
    for (int nc = 0; nc < NC; ++nc) {
        float bs = base[nc * HID + t];
        const float* cf = cfg_feat + ((size_t)c * NC + nc) * CFGD;
        float p = bs;
#pragma unroll
        for (int k = 0; k < CFGD; ++k) p += cf[k] * wc[k];
        p = gelu_f(p);
        p = se_apply(p, t, w1r, w2c, b1t, b2t, psh, part, s1);
        float se = sumexp[(nc * 2 + chunk) * HID + t];
        acc += p * (expf(p * invT) / se);
        __syncthreads();
    }
    cfge[(size_t)c * HID + t] = acc * (1.0f / (float)NC);
}

// ---------------------------------------------------------------------------
// Head MLP: scores[c]
// ---------------------------------------------------------------------------
__global__ void __launch_bounds__(128) k_head(const float* __restrict__ gsum,
                                              const float* __restrict__ cfge,
                                              const float* __restrict__ w1,
                                              const float* __restrict__ b1,
                                              const float* __restrict__ w2,
                                              const float* __restrict__ b2,
                                              const float* __restrict__ w3,
                                              const float* __restrict__ b3,
                                              float* __restrict__ out, int N) {
    __shared__ float fin[2 * HID], x1[HID], x2[64];
    int c = blockIdx.x, t = threadIdx.x;
    fin[t] = gsum[t] * (1.0f / (float)N);
    fin[HID + t] = cfge[(size_t)c * HID + t];
    __syncthreads();
    float a = b1[t];
#pragma unroll 8
    for (int k = 0; k < 2 * HID; ++k) a += fin[k] * w1[k * HID + t];
    x1[t] = gelu_f(a);
    __syncthreads();
    if (t < 64) {
        float a2 = b2[t];
#pragma unroll 8
        for (int k = 0; k < HID; ++k) a2 += x1[k] * w2[k * 64 + t];
        x2[t] = gelu_f(a2);
    }
    __syncthreads();
    if (t == 0) {
        float s = b3[0];
#pragma unroll
        for (int k = 0; k < 64; ++k) s += x2[k] * w3[k];
        out[c] = s;
    }
}

// ---------------------------------------------------------------------------
extern "C" void kernel_launch(void* const* d_in, const int* in_sizes, int n_in,
                              void* d_out, int out_size, void* d_ws, size_t ws_size,
                              hipStream_t stream) {
    const float* node_feat  = (const float*)d_in[0];
    const int*   node_opc   = (const int*)d_in[1];
    const float* topo       = (const float*)d_in[2];
    const int*   edge_index = (const int*)d_in[3];
    const int*   cfg_ids    = (const int*)d_in[4];
    const float* cfg_feat   = (const float*)d_in[5];
    const float* temp_p     = (const float*)d_in[6];
    const float* opc_embed  = (const float*)d_in[7];
    const float* in_w       = (const float*)d_in[8];
    const float* in_b       = (const float*)d_in[9];
    const float* in_g       = (const float*)d_in[10];
    const float* in_beta    = (const float*)d_in[11];
    const float* sage_Wl    = (const float*)d_in[12];
    const float* sage_bl    = (const float*)d_in[13];
    const float* sage_Wr    = (const float*)d_in[14];
    const float* ln_g       = (const float*)d_in[15];
    const float* ln_b       = (const float*)d_in[16];
    const float* cfg_w      = (const float*)d_in[17];
    const float* cfg_b      = (const float*)d_in[18];
    const float* se_w1      = (const float*)d_in[19];
    const float* se_b1      = (const float*)d_in[20];
    const float* se_w2      = (const float*)d_in[21];
    const float* se_b2      = (const float*)d_in[22];
    const float* h_w1       = (const float*)d_in[23];
    const float* h_b1       = (const float*)d_in[24];
    const float* h_w2       = (const float*)d_in[25];
    const float* h_b2       = (const float*)d_in[26];
    const float* h_w3       = (const float*)d_in[27];
    const float* h_b3       = (const float*)d_in[28];
    float* out = (float*)d_out;

    const int N  = in_sizes[1];
    const int M  = in_sizes[3] / 2;
    const int NC = in_sizes[4];
    const int C  = in_sizes[5] / (NC * CFGD);

    // ---- workspace carve (256B aligned) ----
    char* w = (char*)d_ws;
    size_t off = 0;
    auto carve = [&](size_t bytes) -> void* {
        void* p = w + off;
        off += (bytes + 255) & ~(size_t)255;
        return p;
    };
    float*    hbuf     = (float*)carve((size_t)N * HID * 4);
    float*    conv     = (float*)carve((size_t)N * HID * 4);  // aliases agg
    unsigned* csr_ptr  = (unsigned*)carve((size_t)(N + 1) * 4);
    int*      csr_src  = (int*)carve((size_t)M * 4);
    unsigned* deg      = (unsigned*)carve((size_t)N * 4);
    unsigned* cursor   = (unsigned*)carve((size_t)N * 4);
    float*    opc_proj = (float*)carve((size_t)NOPC * HID * 4);
    float*    basev    = (float*)carve((size_t)NC * HID * 4);
    float*    sumexp   = (float*)carve((size_t)NC * 2 * HID * 4);
    float*    cfge     = (float*)carve((size_t)C * HID * 4);
    float*    gsum     = (float*)carve(HID * 4);
    (void)ws_size;

    (void)hipMemsetAsync(deg, 0, (size_t)N * 4, stream);
    (void)hipMemsetAsync(cursor, 0, (size_t)N * 4, stream);
    (void)hipMemsetAsync(gsum, 0, HID * 4, stream);

    const int rowTiles   = (N + 15) / 16;
    const int gemmBlocks = (rowTiles + 7) / 8;
    const int nodeBlocks = (N * 32 + 255) / 256;  // wave per node, 8 waves/block
    const int edgeBlocks = (M + 255) / 256;

    // 1) opcode-projection fold + input embed
    k_opc_proj<<<NOPC, 128, 0, stream>>>(opc_embed, in_w, opc_proj);
    k_embed_gemm<<<gemmBlocks, 256, 0, stream>>>(node_feat, in_w, conv, N);
    k_embed_post<<<nodeBlocks, 256, 0, stream>>>(conv, node_opc, topo, opc_proj, in_w,
                                                 in_b, in_g, in_beta, hbuf, N);

    // 2) CSR build (once; degrees fixed across layers)
    k_deg<<<edgeBlocks, 256, 0, stream>>>(edge_index, deg, M);
    k_scan<<<1, 1024, 0, stream>>>(deg, csr_ptr, N);
    k_fill<<<edgeBlocks, 256, 0, stream>>>(edge_index, csr_ptr, cursor, csr_src, M);

    // 3) SAGE layers
    for (int l = 0; l < 4; ++l) {
        k_agg<<<nodeBlocks, 256, 0, stream>>>(hbuf, csr_ptr, csr_src, conv, N);
        k_sage_gemm<<<gemmBlocks, 256, 0, stream>>>(conv, hbuf,
                                                    sage_Wl + (size_t)l * HID * HID,
                                                    sage_Wr + (size_t)l * HID * HID, conv, N);
        k_sage_post<<<nodeBlocks, 256, 0, stream>>>(hbuf, conv, sage_bl + (size_t)l * HID,
                                                    ln_g + (size_t)l * HID,
                                                    ln_b + (size_t)l * HID, N);
    }

    // 4) global embed + config scoring
    k_mean<<<200, 128, 0, stream>>>(hbuf, gsum, N);
    k_base<<<NC, 128, 0, stream>>>(hbuf, cfg_ids, cfg_w, cfg_b, basev);
    k_pass1<<<NC, 128, 0, stream>>>(basev, cfg_feat, cfg_w, se_w1, se_b1, se_w2, se_b2,
                                    temp_p, sumexp, C, NC);
    k_pass2<<<C, 128, 0, stream>>>(basev, cfg_feat, cfg_w, se_w1, se_b1, se_w2, se_b2,
                                   sumexp, temp_p, cfge, C, NC);

    // 5) head
    k_head<<<C, 128, 0, stream>>>(gsum, cfge, h_w1, h_b1, h_w2, h_b2, h_w3, h_b3, out, N);
    (void)n_in; (void)out_size;
}